// Outer_73787538145889
// MI455X (gfx1250) — compile-verified
//
#include <hip/hip_runtime.h>
#include <hip/hip_bf16.h>

#define H_ 16
#define P_ 192
#define D_ 32

typedef __attribute__((ext_vector_type(2))) float v2f;
typedef __attribute__((ext_vector_type(8))) float v8f;

// ---------------------------------------------------------------------------
// Kernel 1: per-head GEMMs  L[h] = X[h] @ W_left[h],  R[h] = X[h] @ W_right[h]
// via V_WMMA_F32_16X16X4_F32.  grid = 16 heads * 6 tile-groups, 8 waves/block,
// one 16x16 output tile per wave, 48 k-steps of k=4 each.
// ---------------------------------------------------------------------------
__global__ __launch_bounds__(256)
void bilinear_wmma_kernel(const float* __restrict__ X,
                          const float* __restrict__ Wl,
                          const float* __restrict__ Wr,
                          float* __restrict__ L,
                          float* __restrict__ R)
{
    const int wave = threadIdx.x >> 5;   // 0..7 (wave32)
    const int lane = threadIdx.x & 31;
    const int half = lane >> 4;          // 0: lanes 0-15, 1: lanes 16-31
    const int row  = lane & 15;

    const int h    = blockIdx.x / 6;
    const int tile = (blockIdx.x % 6) * 8 + wave;   // 0..47
    const int mat  = tile / 24;                     // 0 = left, 1 = right
    const int rem  = tile % 24;
    const int m0   = (rem >> 1) * 16;               // 0..176
    const int n0   = (rem & 1) * 16;                // 0 or 16

    const float* __restrict__ Xh = X + h * (P_ * P_);
    const float* __restrict__ Wh = (mat == 0 ? Wl : Wr) + h * (P_ * D_);
    float* __restrict__ Out      = (mat == 0 ? L  : R ) + h * (P_ * D_);

    // A fragment: lane(row,half) holds X[m0+row][k0 + 2*half + {0,1}]
    const float* aRow = Xh + (m0 + row) * P_ + 2 * half;
    // B fragment: lane(row,half) holds W[k0 + 2*half + {0,1}][n0+row]
    const float* bCol = Wh + (2 * half) * D_ + n0 + row;

    v8f c = {};
    #pragma unroll 4
    for (int k0 = 0; k0 < P_; k0 += 4) {
        v2f a = *(const v2f*)(aRow + k0);          // contiguous pair -> b64 load
        v2f b;
        b.x = bCol[k0 * D_];
        b.y = bCol[k0 * D_ + D_];
        // D = A(16x4 f32) * B(4x16 f32) + C, exact fp32 matrix pipe
        c = __builtin_amdgcn_wmma_f32_16x16x4_f32(
                /*neg_a=*/false, a, /*neg_b=*/false, b,
                /*c_mod=*/(short)0, c, /*reuse_a=*/false, /*reuse_b=*/false);
    }

    // C/D layout: VGPR r -> M = r + 8*half, N = row
    #pragma unroll
    for (int r = 0; r < 8; ++r) {
        Out[(m0 + r + 8 * half) * D_ + n0 + row] = c[r];
    }
}

// ---------------------------------------------------------------------------
// Kernel 2: cheap reductions.  T[h,a]=sum_t X, Q[h,a]=sum_t X^2,
// S_l[h,d]=sum_b Wl, S_r[h,d]=sum_b Wr.   grid = 16 heads.
// ---------------------------------------------------------------------------
__global__ __launch_bounds__(256)
void sums_kernel(const float* __restrict__ X,
                 const float* __restrict__ Wl,
                 const float* __restrict__ Wr,
                 float* __restrict__ T, float* __restrict__ Q,
                 float* __restrict__ Sl, float* __restrict__ Sr)
{
    const int h   = blockIdx.x;
    const int tid = threadIdx.x;
    if (tid < P_) {
        const float* x = X + (h * P_ + tid) * P_;
        float s = 0.f, q = 0.f;
        #pragma unroll 4
        for (int t = 0; t < P_; ++t) { float v = x[t]; s += v; q += v * v; }
        T[h * P_ + tid] = s;
        Q[h * P_ + tid] = q;
    } else {
        const int r   = tid - P_;      // 0..63
        const int mat = r >> 5;        // 0 = left, 1 = right
        const int d   = r & 31;
        const float* w = (mat == 0 ? Wl : Wr) + h * (P_ * D_) + d;
        float s = 0.f;
        #pragma unroll 4
        for (int b = 0; b < P_; ++b) s += w[b * D_];
        (mat == 0 ? Sl : Sr)[h * D_ + d] = s;
    }
}

// ---------------------------------------------------------------------------
// Kernel 3: X_att[h,a,d] = Sl*Sr*Q - (Sl*R + Sr*L)*T + P*L*R, times fc_w,
// block-reduced deterministically to one partial per block.
// grid = 98304/256 = 384 blocks.
// ---------------------------------------------------------------------------
__global__ __launch_bounds__(256)
void attn_reduce_kernel(const float* __restrict__ L, const float* __restrict__ R,
                        const float* __restrict__ T, const float* __restrict__ Q,
                        const float* __restrict__ Sl, const float* __restrict__ Sr,
                        const float* __restrict__ fcw, float* __restrict__ partials)
{
    __shared__ float sm[256];
    const int idx = blockIdx.x * 256 + threadIdx.x;   // < H*P*D
    const int d   = idx & (D_ - 1);
    const int ha  = idx / D_;
    const int h   = ha / P_;

    const float l  = L[idx],          r  = R[idx];
    const float t  = T[ha],           q  = Q[ha];
    const float sl = Sl[h * D_ + d],  sr = Sr[h * D_ + d];

    const float att = sl * sr * q - (sl * r + sr * l) * t + (float)P_ * l * r;
    sm[threadIdx.x] = fcw[idx] * att;
    __syncthreads();
    #pragma unroll
    for (int s = 128; s > 0; s >>= 1) {
        if (threadIdx.x < s) sm[threadIdx.x] += sm[threadIdx.x + s];
        __syncthreads();
    }
    if (threadIdx.x == 0) partials[blockIdx.x] = sm[0];
}

// ---------------------------------------------------------------------------
// Kernel 4: deterministic final reduction of 384 partials + bias.
// ---------------------------------------------------------------------------
__global__ __launch_bounds__(256)
void final_reduce_kernel(const float* __restrict__ partials,
                         const float* __restrict__ fcb,
                         float* __restrict__ out)
{
    __shared__ float sm[256];
    float v = partials[threadIdx.x];
    if (threadIdx.x < 128) v += partials[threadIdx.x + 256];
    sm[threadIdx.x] = v;
    __syncthreads();
    #pragma unroll
    for (int s = 128; s > 0; s >>= 1) {
        if (threadIdx.x < s) sm[threadIdx.x] += sm[threadIdx.x + s];
        __syncthreads();
    }
    if (threadIdx.x == 0) out[0] = sm[0] + fcb[0];
}

// ---------------------------------------------------------------------------
extern "C" void kernel_launch(void* const* d_in, const int* in_sizes, int n_in,
                              void* d_out, int out_size, void* d_ws, size_t ws_size,
                              hipStream_t stream) {
    (void)in_sizes; (void)n_in; (void)out_size; (void)ws_size;
    const float* X   = (const float*)d_in[0];   // (H,P,P)
    const float* Wl  = (const float*)d_in[1];   // (H,P,D)
    const float* Wr  = (const float*)d_in[2];   // (H,P,D)
    const float* fcw = (const float*)d_in[3];   // (1, H*P*D)
    const float* fcb = (const float*)d_in[4];   // (1,)
    float* out = (float*)d_out;

    float* ws = (float*)d_ws;
    float* L        = ws;                       // H*P*D = 98304
    float* R        = L  + H_ * P_ * D_;        // 98304
    float* T        = R  + H_ * P_ * D_;        // H*P  = 3072
    float* Q        = T  + H_ * P_;             // 3072
    float* Sl       = Q  + H_ * P_;             // H*D  = 512
    float* Sr       = Sl + H_ * D_;             // 512
    float* partials = Sr + H_ * D_;             // 384

    sums_kernel<<<H_, 256, 0, stream>>>(X, Wl, Wr, T, Q, Sl, Sr);
    bilinear_wmma_kernel<<<H_ * 6, 256, 0, stream>>>(X, Wl, Wr, L, R);
    attn_reduce_kernel<<<(H_ * P_ * D_) / 256, 256, 0, stream>>>(L, R, T, Q, Sl, Sr, fcw, partials);
    final_reduce_kernel<<<1, 256, 0, stream>>>(partials, fcb, out);
}